// MMCL_inv_37778532335646
// MI455X (gfx1250) — compile-verified
//
#include <hip/hip_runtime.h>
#include <hip/hip_bf16.h>

typedef __attribute__((ext_vector_type(2))) float v2f;
typedef __attribute__((ext_vector_type(8))) float v8f;

#define BS   384
#define DIM  256
#define LDK  768            // K buffer row stride (384x768)
#define GINV 14.2857142857f // 1/0.07
// DD = 1.1*I + 1*1^T + E, ||E|| ~ 1e-8  ->  2-cluster spectrum; CG converges
// to f32 machine precision in ~3 iterations. 12 gives a 4x margin while
// cutting launch count (the dominant end-to-end cost) by 5x vs 64.
#define CG_ITERS 12

// ---------------------------------------------------------------------------
// Kernel 1: K[384x768] = exp(2g * (F[:384] @ F^T) - 2g)
// One wave (32 lanes) computes a 16x64 strip via 4 accumulators of
// v_wmma_f32_16x16x4_f32.  A[m][k] = F[m][k], B[k][n] = F[n][k] (both
// contiguous in k -> b64 loads per the 16x4 f32 WMMA operand layout:
// lanes 0-15 hold K = kk+{0,1}, lanes 16-31 hold K = kk+{2,3}).
// ---------------------------------------------------------------------------
__global__ void __launch_bounds__(32)
k_sim(const float* __restrict__ feat, float* __restrict__ Kbuf) {
    const int lane  = threadIdx.x;          // 0..31
    const int tileM = blockIdx.x;           // 0..23
    const int n0    = blockIdx.y * 64;      // 0..704
    const int l15   = lane & 15;
    const int kb    = (lane >> 4) * 2;      // 0 or 2

    const int mrow = tileM * 16 + l15;                  // < 384 (view 0)
    const float* __restrict__ arow = feat + mrow * (2 * DIM);
    const float* __restrict__ brow[4];
#pragma unroll
    for (int j = 0; j < 4; ++j) {
        int n = n0 + 16 * j + l15;                      // 0..767
        brow[j] = feat + ((n < BS) ? n * (2 * DIM) : (n - BS) * (2 * DIM) + DIM);
    }

    v8f acc[4] = {};
    for (int kk = 0; kk < DIM; kk += 4) {
        __builtin_prefetch(arow + kk + 64, 0, 3);   // near-scope prefetch
        v2f a = *(const v2f*)(arow + kk + kb);
#pragma unroll
        for (int j = 0; j < 4; ++j) {
            v2f b = *(const v2f*)(brow[j] + kk + kb);
            acc[j] = __builtin_amdgcn_wmma_f32_16x16x4_f32(
                false, a, false, b, (short)0, acc[j], false, false);
        }
    }

    // C/D layout: VGPR r -> M = r (lanes 0-15) or r+8 (lanes 16-31); N = lane&15
    const int rbase = tileM * 16 + (lane >> 4) * 8;
#pragma unroll
    for (int j = 0; j < 4; ++j) {
        const int col = n0 + 16 * j + l15;
#pragma unroll
        for (int r = 0; r < 8; ++r) {
            float s = acc[j][r];
            Kbuf[(rbase + r) * LDK + col] = __expf(2.0f * GINV * s - 2.0f * GINV);
        }
    }
}

// ---------------------------------------------------------------------------
// Kernel 2: Gt[384x384] = Pt @ KK  (KK symmetric -> B[k][n] = Kbuf[n*LDK+k],
// contiguous in k). Same 16x64-strip WMMA structure; 144 full-EXEC waves.
// ---------------------------------------------------------------------------
__global__ void __launch_bounds__(32)
k_gemmG(const float* __restrict__ Kbuf, const float* __restrict__ Pt,
        float* __restrict__ Gt) {
    const int lane  = threadIdx.x;
    const int tileM = blockIdx.x;           // 0..23
    const int n0    = blockIdx.y * 64;      // 0..320
    const int l15   = lane & 15;
    const int kb    = (lane >> 4) * 2;

    const float* __restrict__ arow = Pt + (tileM * 16 + l15) * BS;
    const float* __restrict__ brow[4];
#pragma unroll
    for (int j = 0; j < 4; ++j)
        brow[j] = Kbuf + (n0 + 16 * j + l15) * LDK;    // row n of symmetric KK

    v8f acc[4] = {};
    for (int kk = 0; kk < BS; kk += 4) {
        __builtin_prefetch(arow + kk + 64, 0, 3);
        v2f a = *(const v2f*)(arow + kk + kb);
#pragma unroll
        for (int j = 0; j < 4; ++j) {
            v2f b = *(const v2f*)(brow[j] + kk + kb);
            acc[j] = __builtin_amdgcn_wmma_f32_16x16x4_f32(
                false, a, false, b, (short)0, acc[j], false, false);
        }
    }

    const int rbase = tileM * 16 + (lane >> 4) * 8;
#pragma unroll
    for (int j = 0; j < 4; ++j) {
        const int col = n0 + 16 * j + l15;
#pragma unroll
        for (int r = 0; r < 8; ++r)
            Gt[(rbase + r) * BS + col] = acc[j][r];
    }
}

// ---------------------------------------------------------------------------
// CG state init: R = P = 2*(1 - e_i) per system row, X = 0.
// ---------------------------------------------------------------------------
__global__ void k_init(float* __restrict__ Pt, float* __restrict__ Rt,
                       float* __restrict__ Xt) {
    int idx = blockIdx.x * blockDim.x + threadIdx.x;
    if (idx < BS * BS) {
        int i = idx / BS, a = idx - i * BS;
        float v = (a == i) ? 0.0f : 2.0f;
        Rt[idx] = v; Pt[idx] = v; Xt[idx] = 0.0f;
    }
}

__device__ __forceinline__ float block_reduce128(float v, float* red) {
    int t = threadIdx.x;
    red[t] = v; __syncthreads();
#pragma unroll
    for (int s = 64; s > 0; s >>= 1) {
        if (t < s) red[t] += red[t + s];
        __syncthreads();
    }
    float r = red[0];
    __syncthreads();
    return r;
}

// ---------------------------------------------------------------------------
// CG update (one block = one system i, 128 threads x 3 chunks = 384 coords):
//   q = G_row - KK_row*s - d + s + 0.1*p  (q[i]=0)   [rank-1 corrected matvec]
//   alpha = rr/p'q;  x += alpha p;  r -= alpha q;  beta = rr'/rr;  p = r+beta p
// ---------------------------------------------------------------------------
__global__ void __launch_bounds__(128)
k_cg_update(const float* __restrict__ Kbuf, const float* __restrict__ Gt,
            float* __restrict__ Pt, float* __restrict__ Rt,
            float* __restrict__ Xt) {
    __shared__ float red[128];
    const int i = blockIdx.x, t = threadIdx.x;

    float p[3], r[3], q[3];
    float lp = 0.0f, lrr = 0.0f;
#pragma unroll
    for (int c = 0; c < 3; ++c) {
        int a = t + c * 128;
        p[c] = Pt[i * BS + a];
        r[c] = Rt[i * BS + a];
        lp  += p[c];
        lrr += r[c] * r[c];
    }
    float s  = block_reduce128(lp, red);
    float rr = block_reduce128(lrr, red);
    float d  = Gt[i * BS + i];

    float lpq = 0.0f;
#pragma unroll
    for (int c = 0; c < 3; ++c) {
        int a = t + c * 128;
        float gv = Gt[i * BS + a];
        float kv = Kbuf[i * LDK + a];         // KK row i (symmetric)
        q[c] = gv - kv * s - d + s + 0.1f * p[c];
        if (a == i) q[c] = 0.0f;
        lpq += p[c] * q[c];
    }
    float pap   = block_reduce128(lpq, red);
    float alpha = (pap != 0.0f) ? rr / pap : 0.0f;

    float lrr2 = 0.0f;
#pragma unroll
    for (int c = 0; c < 3; ++c) {
        int a = t + c * 128;
        Xt[i * BS + a] += alpha * p[c];
        r[c] -= alpha * q[c];
        lrr2 += r[c] * r[c];
    }
    float rr2  = block_reduce128(lrr2, red);
    float beta = (rr > 0.0f) ? rr2 / rr : 0.0f;
#pragma unroll
    for (int c = 0; c < 3; ++c) {
        int a = t + c * 128;
        Rt[i * BS + a] = r[c];
        Pt[i * BS + a] = r[c] + beta * p[c];
    }
}

// ---------------------------------------------------------------------------
// loss = (1/384) * sum_i sum_{a!=i} clip(X[i][a],0,1) * (Ks[a,i] - Ks[i,i])
// ---------------------------------------------------------------------------
__global__ void __launch_bounds__(128)
k_loss1(const float* __restrict__ Kbuf, const float* __restrict__ Xt,
        float* __restrict__ partial) {
    __shared__ float red[128];
    const int i = blockIdx.x, t = threadIdx.x;
    const float ksii = Kbuf[i * LDK + BS + i];
    float l = 0.0f;
#pragma unroll
    for (int c = 0; c < 3; ++c) {
        int a = t + c * 128;
        if (a != i) {
            float x = Xt[i * BS + a];
            x = fminf(fmaxf(x, 0.0f), 1.0f);
            float ks = Kbuf[a * LDK + BS + i];
            l += x * (ks - ksii);
        }
    }
    float sum = block_reduce128(l, red);
    if (t == 0) partial[i] = sum;
}

__global__ void __launch_bounds__(128)
k_loss2(const float* __restrict__ partial, float* __restrict__ out) {
    __shared__ float red[128];
    int t = threadIdx.x;
    float v = partial[t] + partial[t + 128] + partial[t + 256];
    float sum = block_reduce128(v, red);
    if (t == 0) out[0] = sum * (1.0f / (float)BS);
}

// ---------------------------------------------------------------------------
extern "C" void kernel_launch(void* const* d_in, const int* in_sizes, int n_in,
                              void* d_out, int out_size, void* d_ws, size_t ws_size,
                              hipStream_t stream) {
    (void)in_sizes; (void)n_in; (void)out_size; (void)ws_size;
    const float* feat = (const float*)d_in[0];       // (384, 2, 256) f32

    float* ws      = (float*)d_ws;
    float* Kbuf    = ws;                  // 384*768
    float* Pt      = Kbuf + BS * LDK;     // 384*384
    float* Rt      = Pt + BS * BS;
    float* Xt      = Rt + BS * BS;
    float* Gt      = Xt + BS * BS;
    float* partial = Gt + BS * BS;        // 384

    k_sim<<<dim3(24, 12), 32, 0, stream>>>(feat, Kbuf);
    k_init<<<(BS * BS + 255) / 256, 256, 0, stream>>>(Pt, Rt, Xt);

    for (int it = 0; it < CG_ITERS; ++it) {
        k_gemmG<<<dim3(24, 6), 32, 0, stream>>>(Kbuf, Pt, Gt);
        k_cg_update<<<BS, 128, 0, stream>>>(Kbuf, Gt, Pt, Rt, Xt);
    }

    k_loss1<<<BS, 128, 0, stream>>>(Kbuf, Xt, partial);
    k_loss2<<<1, 128, 0, stream>>>(partial, (float*)d_out);
}